// SopaEncoder_79637283602655
// MI455X (gfx1250) — compile-verified
//
#include <hip/hip_runtime.h>
#include <hip/hip_bf16.h>
#include <math.h>

typedef __attribute__((ext_vector_type(2))) float v2f;
typedef __attribute__((ext_vector_type(8))) float v8f;

#define L_SEQ 256
#define B_SZ  128
#define E_SZ  256
#define H_SZ  256
#define P_SZ  100
#define M_SZ  6
#define G4H   1024
#define NEGV  (-100.0f)

__device__ __forceinline__ v8f wmma_f32(v2f a, v2f b, v8f c) {
  // V_WMMA_F32_16X16X4_F32 : D = A(16x4) x B(4x16) + C(16x16), all f32
  return __builtin_amdgcn_wmma_f32_16x16x4_f32(
      /*neg_a=*/false, a, /*neg_b=*/false, b,
      /*c_mod=*/(short)0, c, /*reuse_a=*/false, /*reuse_b=*/false);
}

__device__ __forceinline__ float sigm(float x) {
  return 1.0f / (1.0f + __expf(-x));
}

// ---------------------------------------------------------------------------
// Embedding gather: emb[(l*B+b)*E + :] = table[input[l*B+b]*E + :]
// ---------------------------------------------------------------------------
__global__ void gather_emb(const int* __restrict__ idx,
                           const float* __restrict__ table,
                           float* __restrict__ emb) {
  int g = blockIdx.x * blockDim.x + threadIdx.x;   // row*64 + chunk
  if (g >= (L_SEQ * B_SZ) * (E_SZ / 4)) return;
  int row = g >> 6;
  int c4  = g & 63;
  int tok = idx[row];
  const float4* src = (const float4*)(table + (size_t)tok * E_SZ);
  float4* dst = (float4*)(emb + (size_t)row * E_SZ);
  dst[c4] = src[c4];
}

// ---------------------------------------------------------------------------
// WMMA GEMM:  C[m][n] = sum_k A[m*K+k] * (Bm[n*K+k] (+ B2[n*K+k]))
//                        (+ biasM[m]) (+ biasN[n])
// One wave computes a 16x64 C strip (4 accumulator tiles sharing A).
// K-loop is chunked: 8 K-steps (40 b64 loads) issued before a 32-WMMA burst.
// __launch_bounds__(256,1): allow the full VGPR budget so the ~112 live
// operand registers do NOT spill to scratch inside the hot loop.
// Layouts per ISA 7.12.2 (f32 16x16x4):
//   A: lane=16*kh+m, vgpr0/1 = A[m][2kh+0/1]
//   B: lane=16*kh+n, vgpr0/1 = B[2kh+0/1][n]
//   C: vgpr r, lane=16*kh+n -> C[r+8*kh][n]
// ---------------------------------------------------------------------------
__global__ __launch_bounds__(256, 1) void gemm_tn_wmma(
                             const float* __restrict__ A,
                             const float* __restrict__ Bm,
                             const float* __restrict__ B2,
                             const float* __restrict__ biasM,
                             const float* __restrict__ biasN,
                             float* __restrict__ C,
                             int Mt, int Ng, int K, long ldc)
{
  int lane = threadIdx.x & 31;
  int wave = threadIdx.x >> 5;
  int task = blockIdx.x * (blockDim.x >> 5) + wave;
  if (task >= Mt * Ng) return;                 // wave-uniform: EXEC stays full
  int mt = task / Ng, ng = task - mt * Ng;
  int m0 = mt << 4, n0 = ng << 6;              // 64-wide N strip
  int lm = lane & 15, kh = lane >> 4;

  const float* arow = A  + (size_t)(m0 + lm) * K + 2 * kh;
  const float* br   = Bm + (size_t)(n0 + lm) * K + 2 * kh;
  const size_t gK = (size_t)16 * K;            // next 16-column tile of B

  v8f ac0 = {}, ac1 = {}, ac2 = {}, ac3 = {};

  for (int k = 0; k < K; k += 32) {            // 8 K-steps per chunk
    v2f a[8], b0[8], b1[8], b2[8], b3[8];
#pragma unroll
    for (int i = 0; i < 8; ++i) {
      a[i]  = *(const v2f*)(arow + k + 4 * i);
      b0[i] = *(const v2f*)(br            + k + 4 * i);
      b1[i] = *(const v2f*)(br + gK       + k + 4 * i);
      b2[i] = *(const v2f*)(br + 2 * gK   + k + 4 * i);
      b3[i] = *(const v2f*)(br + 3 * gK   + k + 4 * i);
    }
    if (B2) {
      const float* cr = B2 + (size_t)(n0 + lm) * K + 2 * kh;
#pragma unroll
      for (int i = 0; i < 8; ++i) {
        b0[i] += *(const v2f*)(cr          + k + 4 * i);
        b1[i] += *(const v2f*)(cr + gK     + k + 4 * i);
        b2[i] += *(const v2f*)(cr + 2 * gK + k + 4 * i);
        b3[i] += *(const v2f*)(cr + 3 * gK + k + 4 * i);
      }
    }
#pragma unroll
    for (int i = 0; i < 8; ++i) {
      ac0 = wmma_f32(a[i], b0[i], ac0);
      ac1 = wmma_f32(a[i], b1[i], ac1);
      ac2 = wmma_f32(a[i], b2[i], ac2);
      ac3 = wmma_f32(a[i], b3[i], ac3);
    }
  }

  float bm[8];
#pragma unroll
  for (int r = 0; r < 8; ++r)
    bm[r] = biasM ? biasM[m0 + r + 8 * kh] : 0.0f;

#pragma unroll
  for (int g = 0; g < 4; ++g) {
    v8f acc = (g == 0) ? ac0 : (g == 1) ? ac1 : (g == 2) ? ac2 : ac3;
    int ncol = n0 + g * 16 + lm;
    float bn = biasN ? biasN[ncol] : 0.0f;
#pragma unroll
    for (int r = 0; r < 8; ++r) {
      int mrow = m0 + r + 8 * kh;
      C[(size_t)mrow * ldc + ncol] = acc[r] + bm[r] + bn;
    }
  }
}

// ---------------------------------------------------------------------------
// Persistent bidirectional LSTM: block 0 = forward, block 1 = backward.
// One 512-thread workgroup (16 waves, 4 waves/SIMD -> double VGPR headroom,
// no spills) per direction; each wave handles 8 of the 128 (b,j) tile tasks.
// Serial loop over L with workgroup barriers; h ping-pong in global
// (WGP$-resident), c owned per wave. Per step:
//   G(128x1024) = xg[t] + h @ Whh^T  via f32 WMMA (K chunked by 4 steps:
//   20 b64 loads in flight per 16-WMMA burst), then fused gates.
// ---------------------------------------------------------------------------
__global__ __launch_bounds__(512, 1) void lstm_wmma(
    const float* __restrict__ xg_f, const float* __restrict__ xg_b,
    const float* __restrict__ Whh_f, const float* __restrict__ Whh_b,
    float* __restrict__ out_f, float* __restrict__ out_b,
    float* __restrict__ hbuf, float* __restrict__ cbuf)
{
  const int dir = blockIdx.x;
  const float* xg  = dir ? xg_b  : xg_f;
  const float* W   = dir ? Whh_b : Whh_f;   // (4H, H) row-major == B[n*K+k]
  float* out = dir ? out_b : out_f;
  float* hb = hbuf + (size_t)dir * 2 * B_SZ * H_SZ;   // ping-pong
  float* cb = cbuf + (size_t)dir * B_SZ * H_SZ;

  const int tid = threadIdx.x;
  for (int i = tid; i < B_SZ * H_SZ; i += 512) { hb[i] = 0.0f; cb[i] = 0.0f; }
  __threadfence_block();
  __syncthreads();

  const int lane = tid & 31, wave = tid >> 5;
  const int lm = lane & 15, kh = lane >> 4;

  for (int t = 0; t < L_SEQ; ++t) {
    const int ts = dir ? (L_SEQ - 1 - t) : t;
    const float* hp = hb + (t & 1) * B_SZ * H_SZ;
    float*       hn = hb + ((t + 1) & 1) * B_SZ * H_SZ;
    const float* xrow = xg + (size_t)ts * B_SZ * G4H;

    for (int q = 0; q < 8; ++q) {
      const int task = wave * 8 + q;           // 0..127 = 8 b-tiles x 16 j-tiles
      const int b0 = (task >> 4) * 16;
      const int j0 = (task & 15) * 16;

      v8f a0, a1, a2, a3;                      // i, f, g, o gate tiles
#pragma unroll
      for (int r = 0; r < 8; ++r) {            // seed accumulators with xg + b
        const size_t xi = (size_t)(b0 + r + 8 * kh) * G4H + j0 + lm;
        a0[r] = xrow[xi];
        a1[r] = xrow[xi + 256];
        a2[r] = xrow[xi + 512];
        a3[r] = xrow[xi + 768];
      }

      const float* arow = hp + (b0 + lm) * H_SZ + 2 * kh;
      const float* w0   = W + (size_t)(j0 + lm) * H_SZ + 2 * kh;
      for (int k = 0; k < H_SZ; k += 16) {     // 4 K-steps per chunk
        v2f av[4], wv0[4], wv1[4], wv2[4], wv3[4];
#pragma unroll
        for (int i = 0; i < 4; ++i) {
          av[i]  = *(const v2f*)(arow        + k + 4 * i);
          wv0[i] = *(const v2f*)(w0          + k + 4 * i);
          wv1[i] = *(const v2f*)(w0 +  65536 + k + 4 * i);
          wv2[i] = *(const v2f*)(w0 + 131072 + k + 4 * i);
          wv3[i] = *(const v2f*)(w0 + 196608 + k + 4 * i);
        }
#pragma unroll
        for (int i = 0; i < 4; ++i) {
          a0 = wmma_f32(av[i], wv0[i], a0);
          a1 = wmma_f32(av[i], wv1[i], a1);
          a2 = wmma_f32(av[i], wv2[i], a2);
          a3 = wmma_f32(av[i], wv3[i], a3);
        }
      }

#pragma unroll
      for (int r = 0; r < 8; ++r) {
        const int bb = b0 + r + 8 * kh;
        const int j  = j0 + lm;
        const int ci = bb * H_SZ + j;
        const float cprev = cb[ci];
        const float cn = sigm(a1[r]) * cprev + sigm(a0[r]) * tanhf(a2[r]);
        cb[ci] = cn;
        const float h = sigm(a3[r]) * tanhf(cn);
        hn[ci] = h;
        out[(size_t)ts * B_SZ * H_SZ + ci] = h;
      }
    }
    __threadfence_block();
    __syncthreads();
  }
}

// ---------------------------------------------------------------------------
// SoPa DP scan. One lane per (b, p); 6 states in registers; 256 serial steps.
// The reference's trans reshape is a raw reinterpretation of the row-major
// (1200 x 32768) sc buffer: flat index = b*307200 + t*1200 + p*12 + s*6 + m.
// ---------------------------------------------------------------------------
__global__ void sopa_dp(const float* __restrict__ sc,
                        const float* __restrict__ eps,
                        const float* __restrict__ slsp,
                        const int* __restrict__ ilen,
                        float* __restrict__ out)
{
  int g = blockIdx.x * blockDim.x + threadIdx.x;
  if (g >= B_SZ * P_SZ) return;
  int b = g / P_SZ, p = g - b * P_SZ;

  float e[5];
#pragma unroll
  for (int i = 0; i < 5; ++i) e[i] = eps[p * 5 + i];
  const float sls = slsp[0];
  const bool end4 = (p < 50);                 // END_STATES: 4 (p<50) else 5
  const int len = ilen[b];

  float hid[6];
  hid[0] = 0.0f;
#pragma unroll
  for (int m = 1; m < 6; ++m) hid[m] = NEGV;
  float score = NEGV;

  const float* base = sc + (size_t)b * (L_SEQ * P_SZ * 2 * M_SZ) + p * 12;
  for (int t = 0; t < L_SEQ; ++t) {
    const float4* q4 = (const float4*)(base + (size_t)t * (P_SZ * 2 * M_SZ));
    float4 q0 = q4[0], q1 = q4[1], q2 = q4[2];
    float tr0[6] = {q0.x, q0.y, q0.z, q0.w, q1.x, q1.y};   // self-loop row
    float tr1[6] = {q1.z, q1.w, q2.x, q2.y, q2.z, q2.w};   // main-path row

    float ae[6];
    ae[0] = fmaxf(hid[0], NEGV);
#pragma unroll
    for (int m = 1; m < 6; ++m) ae[m] = fmaxf(hid[m], hid[m - 1] + e[m - 1]);

    hid[0] = fmaxf(0.0f, sls + ae[0] + tr0[0]);
#pragma unroll
    for (int m = 1; m < 6; ++m)
      hid[m] = fmaxf(ae[m - 1] + tr1[m - 1], sls + ae[m] + tr0[m]);

    float ev = end4 ? hid[4] : hid[5];
    if (t < len) score = fmaxf(score, ev);
  }
  out[b * P_SZ + p] = score;
}

// ---------------------------------------------------------------------------
extern "C" void kernel_launch(void* const* d_in, const int* in_sizes, int n_in,
                              void* d_out, int out_size, void* d_ws, size_t ws_size,
                              hipStream_t stream) {
  (void)in_sizes; (void)n_in; (void)out_size; (void)ws_size;
  const int*   input   = (const int*)d_in[0];
  const int*   ilen    = (const int*)d_in[1];
  const float* table   = (const float*)d_in[2];
  const float* Wih_f   = (const float*)d_in[3];
  const float* Whh_f   = (const float*)d_in[4];
  const float* b_f     = (const float*)d_in[5];
  const float* Wih_b   = (const float*)d_in[6];
  const float* Whh_b   = (const float*)d_in[7];
  const float* b_b     = (const float*)d_in[8];
  const float* diags   = (const float*)d_in[9];
  const float* bias    = (const float*)d_in[10];
  const float* eps     = (const float*)d_in[11];
  const float* sls     = (const float*)d_in[12];
  float* out = (float*)d_out;

  float* ws = (float*)d_ws;
  const size_t LB = (size_t)L_SEQ * B_SZ;            // 32768
  float* emb  = ws;                                  //  8,388,608 f
  float* xg_f = emb  + LB * E_SZ;                    // 33,554,432 f
  float* xg_b = xg_f + LB * G4H;                     // 33,554,432 f
  float* outf = xg_b + LB * G4H;                     //  8,388,608 f
  float* outb = outf + LB * H_SZ;                    //  8,388,608 f
  float* hbuf = outb + LB * H_SZ;                    //    131,072 f
  float* cbuf = hbuf + 4 * B_SZ * H_SZ;              //     65,536 f
  float* sc   = xg_f;   // alias: 39,321,600 f fits in the dead 67M-f xg region

  // 1) embedding gather
  gather_emb<<<(int)((LB * (E_SZ / 4)) / 256), 256, 0, stream>>>(input, table, emb);
  // 2) xg = emb @ Wih^T + b  (both directions)   M=2048 tiles, N=16 strips, K=256
  gemm_tn_wmma<<<(2048 * 16) / 8, 256, 0, stream>>>(emb, Wih_f, nullptr, nullptr, b_f,
                                                    xg_f, 2048, 16, E_SZ, G4H);
  gemm_tn_wmma<<<(2048 * 16) / 8, 256, 0, stream>>>(emb, Wih_b, nullptr, nullptr, b_b,
                                                    xg_b, 2048, 16, E_SZ, G4H);
  // 3) serial bidirectional LSTM (2 persistent workgroups)
  lstm_wmma<<<2, 512, 0, stream>>>(xg_f, xg_b, Whh_f, Whh_b, outf, outb, hbuf, cbuf);
  // 4) sc = diags @ (out_f + out_b)^T + bias     M=75 tiles, N=512 strips, K=256
  gemm_tn_wmma<<<(75 * 512) / 8, 256, 0, stream>>>(diags, outf, outb, bias, nullptr,
                                                   sc, 75, 512, H_SZ, (long)LB);
  // 5) max-plus DP scan -> scores (B, P)
  sopa_dp<<<50, 256, 0, stream>>>(sc, eps, sls, ilen, out);
}